// ChannelAttention_88862873354868
// MI455X (gfx1250) — compile-verified
//
#include <hip/hip_runtime.h>

typedef __attribute__((ext_vector_type(2))) float v2f;
typedef __attribute__((ext_vector_type(4))) float v4f;
typedef __attribute__((ext_vector_type(8))) float v8f;

#define BATCH   2
#define CH      64
#define BC      128                  // BATCH*CH channels total
#define NPC     (64 * 128 * 128)    // elements per channel = 1048576
#define SPLITS  64                   // slices per channel in stage 1
#define SLICE   (NPC / SPLITS)       // 16384 floats per block
#define T1      256                  // stage-1 block size (8 waves)
#define T3      256                  // stage-3 block size
#define EPB3    8192                 // elements per stage-3 block (NPC/EPB3 = 128)

__device__ __forceinline__ void top4_insert(float v, float& t0, float& t1,
                                            float& t2, float& t3) {
    if (v > t3) {
        if (v > t0)      { t3 = t2; t2 = t1; t1 = t0; t0 = v; }
        else if (v > t1) { t3 = t2; t2 = t1; t1 = v; }
        else if (v > t2) { t3 = t2; t2 = v; }
        else             { t3 = v; }
    }
}

// ---------------------------------------------------------------------------
// Stage 1: per channel-slice partial sum + partial top-4. 128 ch * 64 splits.
// ---------------------------------------------------------------------------
__global__ void ca_stage1_reduce(const float* __restrict__ x,
                                 float* __restrict__ psum,
                                 float* __restrict__ ptop4) {
    const int ch    = blockIdx.x / SPLITS;
    const int split = blockIdx.x % SPLITS;
    const v4f* xv = (const v4f*)(x + (long long)ch * NPC + (long long)split * SLICE);

    float sum = 0.0f;
    float t0 = -3.4e38f, t1 = -3.4e38f, t2 = -3.4e38f, t3 = -3.4e38f;

    #pragma unroll
    for (int i = 0; i < SLICE / 4 / T1; ++i) {            // 16 iterations of b128
        v4f v = __builtin_nontemporal_load(&xv[i * T1 + threadIdx.x]);
        sum += v.x + v.y + v.z + v.w;
        top4_insert(v.x, t0, t1, t2, t3);
        top4_insert(v.y, t0, t1, t2, t3);
        top4_insert(v.z, t0, t1, t2, t3);
        top4_insert(v.w, t0, t1, t2, t3);
    }

    // wave32 butterfly reduce (disjoint spans each step -> no duplication)
    #pragma unroll
    for (int off = 16; off >= 1; off >>= 1) {
        sum += __shfl_xor(sum, off, 32);
        float o0 = __shfl_xor(t0, off, 32);
        float o1 = __shfl_xor(t1, off, 32);
        float o2 = __shfl_xor(t2, off, 32);
        float o3 = __shfl_xor(t3, off, 32);
        top4_insert(o0, t0, t1, t2, t3);
        top4_insert(o1, t0, t1, t2, t3);
        top4_insert(o2, t0, t1, t2, t3);
        top4_insert(o3, t0, t1, t2, t3);
    }

    __shared__ float s_sum[T1 / 32];
    __shared__ float s_t[T1 / 32][4];
    const int wid = threadIdx.x >> 5;
    if ((threadIdx.x & 31) == 0) {
        s_sum[wid] = sum;
        s_t[wid][0] = t0; s_t[wid][1] = t1; s_t[wid][2] = t2; s_t[wid][3] = t3;
    }
    __syncthreads();
    if (threadIdx.x == 0) {
        float bs = 0.0f;
        float b0 = -3.4e38f, b1 = -3.4e38f, b2 = -3.4e38f, b3 = -3.4e38f;
        #pragma unroll
        for (int w = 0; w < T1 / 32; ++w) {
            bs += s_sum[w];
            top4_insert(s_t[w][0], b0, b1, b2, b3);
            top4_insert(s_t[w][1], b0, b1, b2, b3);
            top4_insert(s_t[w][2], b0, b1, b2, b3);
            top4_insert(s_t[w][3], b0, b1, b2, b3);
        }
        psum[blockIdx.x] = bs;
        float* p = &ptop4[(long long)blockIdx.x * 4];
        p[0] = b0; p[1] = b1; p[2] = b2; p[3] = b3;
    }
}

// ---------------------------------------------------------------------------
// Stage 2: one wave. Finalize avg/top4sum, then MLP via V_WMMA_F32_16X16X4_F32.
// A = 16x64 feature matrix: row0=avg(b0) row1=avg(b1) row2=top4(b0) row3=top4(b1),
// rows 4..15 zero. Layer1: [16,64]x[64,32]; Layer2: [16,32]x[32,64]. Exact f32.
// ---------------------------------------------------------------------------
__global__ void ca_stage2_mlp(const float* __restrict__ psum,
                              const float* __restrict__ ptop4,
                              const float* __restrict__ W1,
                              const float* __restrict__ b1,
                              const float* __restrict__ W2,
                              const float* __restrict__ b2,
                              float* __restrict__ gains,
                              float* __restrict__ out_tail) {
    __shared__ float feat[16][64];   // A-matrix, zero padded
    __shared__ float hmat[16][32];   // hidden activations

    const int lane = threadIdx.x;    // 0..31, single full wave (EXEC all 1s)

    for (int i = lane; i < 16 * 64; i += 32) ((float*)feat)[i] = 0.0f;
    __syncthreads();

    // finalize 4 channels per lane
    for (int j = 0; j < 4; ++j) {
        const int ch = lane + 32 * j;       // 0..127
        const int bb = ch >> 6;             // batch
        const int cc = ch & 63;             // channel
        float s = 0.0f;
        float t0 = -3.4e38f, t1 = -3.4e38f, t2 = -3.4e38f, t3 = -3.4e38f;
        for (int sp = 0; sp < SPLITS; ++sp) {
            s += psum[ch * SPLITS + sp];
            const float* p = &ptop4[(ch * SPLITS + sp) * 4];
            top4_insert(p[0], t0, t1, t2, t3);
            top4_insert(p[1], t0, t1, t2, t3);
            top4_insert(p[2], t0, t1, t2, t3);
            top4_insert(p[3], t0, t1, t2, t3);
        }
        feat[0 + bb][cc] = s * (1.0f / (float)NPC);  // avg
        feat[2 + bb][cc] = t0 + t1 + t2 + t3;        // top-4 sum
    }
    __syncthreads();

    const int mrow  = lane & 15;            // M (A) / N (B,C,D) index
    const int khalf = (lane >> 4) << 1;     // lanes 16..31 hold K+2

    // ---- Layer 1: hidden[16,32] = ReLU(feat @ W1^T + b1); W1^T[k][n]=W1[n*64+k]
    for (int nt = 0; nt < 2; ++nt) {
        v8f acc = {0, 0, 0, 0, 0, 0, 0, 0};
        #pragma unroll
        for (int kk = 0; kk < 16; ++kk) {
            const int k = kk * 4 + khalf;
            const int n = nt * 16 + mrow;
            v2f a, bf;
            a.x  = feat[mrow][k];
            a.y  = feat[mrow][k + 1];
            bf.x = W1[n * 64 + k];
            bf.y = W1[n * 64 + k + 1];
            acc = __builtin_amdgcn_wmma_f32_16x16x4_f32(
                false, a, false, bf, (short)0, acc, false, false);
        }
        const float bias = b1[nt * 16 + mrow];
        #pragma unroll
        for (int v = 0; v < 8; ++v) {
            const int row = v + ((lane >> 4) << 3);    // D layout: vgpr v -> M
            float h = acc[v] + bias;
            hmat[row][nt * 16 + mrow] = h > 0.0f ? h : 0.0f;
        }
    }
    __syncthreads();

    // ---- Layer 2: score[16,64] = hidden @ W2^T + b2; W2^T[k][n]=W2[n*32+k]
    for (int nt = 0; nt < 4; ++nt) {
        v8f acc = {0, 0, 0, 0, 0, 0, 0, 0};
        #pragma unroll
        for (int kk = 0; kk < 8; ++kk) {
            const int k = kk * 4 + khalf;
            const int n = nt * 16 + mrow;
            v2f a, bf;
            a.x  = hmat[mrow][k];
            a.y  = hmat[mrow][k + 1];
            bf.x = W2[n * 32 + k];
            bf.y = W2[n * 32 + k + 1];
            acc = __builtin_amdgcn_wmma_f32_16x16x4_f32(
                false, a, false, bf, (short)0, acc, false, false);
        }
        // D layout: lanes 0..15 hold M=0..7 in vgprs 0..7; rows 0..3 are what we need
        if (lane < 16) {
            const int c  = nt * 16 + lane;
            const float tb = 2.0f * b2[c];              // b2 appears in both fc() calls
            const float s0 = acc[0] + acc[2] + tb;      // batch 0: avg + topk scores
            const float s1 = acc[1] + acc[3] + tb;      // batch 1
            const float g0 = 1.0f / (1.0f + __expf(-s0));
            const float g1 = 1.0f / (1.0f + __expf(-s1));
            gains[c]       = g0;  out_tail[c]       = g0;
            gains[64 + c]  = g1;  out_tail[64 + c]  = g1;
        }
    }
}

// ---------------------------------------------------------------------------
// Stage 3: scaled = x * gain[channel], pure streaming at HBM roofline.
// ---------------------------------------------------------------------------
__global__ void ca_stage3_scale(const float* __restrict__ x,
                                const float* __restrict__ gains,
                                float* __restrict__ out) {
    const long long base = (long long)blockIdx.x * EPB3;
    const int bc = blockIdx.x >> 7;          // NPC/EPB3 == 128 blocks per channel
    const float g = gains[bc];               // uniform -> scalar load
    const v4f* xv = (const v4f*)(x + base);
    v4f* ov = (v4f*)(out + base);
    #pragma unroll
    for (int i = 0; i < EPB3 / 4 / T3; ++i) {            // 8 iterations of b128
        const int idx = i * T3 + threadIdx.x;
        v4f v = __builtin_nontemporal_load(&xv[idx]);
        v *= g;
        __builtin_nontemporal_store(v, &ov[idx]);
    }
}

extern "C" void kernel_launch(void* const* d_in, const int* in_sizes, int n_in,
                              void* d_out, int out_size, void* d_ws, size_t ws_size,
                              hipStream_t stream) {
    (void)in_sizes; (void)n_in; (void)out_size; (void)ws_size;
    const float* x  = (const float*)d_in[0];
    const float* W1 = (const float*)d_in[1];
    const float* b1 = (const float*)d_in[2];
    const float* W2 = (const float*)d_in[3];
    const float* b2 = (const float*)d_in[4];
    float* out = (float*)d_out;

    float* ws_f  = (float*)d_ws;
    float* psum  = ws_f;                      // BC*SPLITS          = 8192 floats
    float* ptop4 = ws_f + BC * SPLITS;        // BC*SPLITS*4        = 32768 floats
    float* gains = ws_f + BC * SPLITS * 5;    // BC                 = 128 floats

    ca_stage1_reduce<<<BC * SPLITS, T1, 0, stream>>>(x, psum, ptop4);
    ca_stage2_mlp<<<1, 32, 0, stream>>>(psum, ptop4, W1, b1, W2, b2,
                                        gains, out + (long long)BC * NPC);
    ca_stage3_scale<<<BC * (NPC / EPB3), T3, 0, stream>>>(x, gains, out);
}